// CoarseMatching_61065845015072
// MI455X (gfx1250) — compile-verified
//
#include <hip/hip_runtime.h>
#include <math.h>

typedef float v2f __attribute__((ext_vector_type(2)));
typedef float v8f __attribute__((ext_vector_type(8)));

#define NDIM 6400          // l and s extent (80*80)
#define CDIM 256           // feature dim
#define HW   80
#define MARGIN 2
#define SCALE 25.6f        // c * TEMP = 256 * 0.1
#define THRESH 0.2f
#define RCH 100            // rows per column-partial chunk
#define NCH 64             // number of chunks (64*100 = 6400)

// ---------------------------------------------------------------------------
// 1) sim = x0 @ x1^T / 25.6 via V_WMMA_F32_16X16X4_F32
//    Block = 256 threads = 8 waves, waves tiled 4(M) x 2(N), 32x32 per wave
//    -> block tile 128x64. Grid: (6400/64, 6400/128) = (100, 50).
// ---------------------------------------------------------------------------
__global__ __launch_bounds__(256)
void gemm_sim_kernel(const float* __restrict__ x0, const float* __restrict__ x1,
                     float* __restrict__ sim) {
    const int lane   = threadIdx.x & 31;
    const int wave   = threadIdx.x >> 5;   // 0..7
    const int waveM  = wave >> 1;          // 0..3
    const int waveN  = wave & 1;           // 0..1
    const int laneLo = lane & 15;
    const int laneHi = lane >> 4;          // 0 or 1

    const int rowBase = blockIdx.y * 128 + waveM * 32;
    const int colBase = blockIdx.x * 64  + waveN * 32;

    // A (16x4 f32) layout: lanes 0-15 = M, VGPR0/1 = K0/K1; lanes 16-31 = K2/K3
    const float* a0 = x0 + (size_t)(rowBase + laneLo) * CDIM;
    const float* a1 = a0 + (size_t)16 * CDIM;
    const float* b0 = x1 + (size_t)(colBase + laneLo) * CDIM;
    const float* b1 = b0 + (size_t)16 * CDIM;

    v8f acc[2][2] = {};

    for (int k = 0; k < CDIM; k += 4) {
        const int kk = k + (laneHi << 1);
        v2f afr[2], bfr[2];
        afr[0] = *(const v2f*)(a0 + kk);
        afr[1] = *(const v2f*)(a1 + kk);
        bfr[0] = *(const v2f*)(b0 + kk);
        bfr[1] = *(const v2f*)(b1 + kk);
#pragma unroll
        for (int i = 0; i < 2; ++i)
#pragma unroll
            for (int j = 0; j < 2; ++j)
                acc[i][j] = __builtin_amdgcn_wmma_f32_16x16x4_f32(
                    false, afr[i], false, bfr[j],
                    (short)0, acc[i][j], false, false);
    }

    // C/D layout: VGPR r -> M = r (lanes 0-15) / r+8 (lanes 16-31), N = lane&15
#pragma unroll
    for (int i = 0; i < 2; ++i) {
#pragma unroll
        for (int j = 0; j < 2; ++j) {
#pragma unroll
            for (int r = 0; r < 8; ++r) {
                const int row = rowBase + i * 16 + r + laneHi * 8;
                const int col = colBase + j * 16 + laneLo;
                sim[(size_t)row * NDIM + col] = acc[i][j][r] / SCALE;
            }
        }
    }
}

// ---------------------------------------------------------------------------
// 2) per-row max and sum(exp(v - max)); one block per row
// ---------------------------------------------------------------------------
__global__ __launch_bounds__(256)
void row_stats_kernel(const float* __restrict__ sim,
                      float* __restrict__ rowmax, float* __restrict__ rowsum) {
    __shared__ float red[256];
    const int l = blockIdx.x;
    const float* row = sim + (size_t)l * NDIM;

    float m = -INFINITY;
    for (int s = threadIdx.x; s < NDIM; s += 256) m = fmaxf(m, row[s]);
    red[threadIdx.x] = m;
    __syncthreads();
    for (int off = 128; off > 0; off >>= 1) {
        if (threadIdx.x < off)
            red[threadIdx.x] = fmaxf(red[threadIdx.x], red[threadIdx.x + off]);
        __syncthreads();
    }
    m = red[0];
    __syncthreads();

    float sum = 0.f;
    for (int s = threadIdx.x; s < NDIM; s += 256) sum += expf(row[s] - m);
    red[threadIdx.x] = sum;
    __syncthreads();
    for (int off = 128; off > 0; off >>= 1) {
        if (threadIdx.x < off) red[threadIdx.x] += red[threadIdx.x + off];
        __syncthreads();
    }
    if (threadIdx.x == 0) { rowmax[l] = m; rowsum[l] = red[0]; }
}

// ---------------------------------------------------------------------------
// 3) column reductions (coalesced, two-stage, deterministic)
// ---------------------------------------------------------------------------
__global__ __launch_bounds__(256)
void col_pmax_kernel(const float* __restrict__ buf, float* __restrict__ part) {
    const int s  = blockIdx.x * 256 + threadIdx.x;
    const int r0 = blockIdx.y * RCH;
    float m = -INFINITY;
    for (int r = 0; r < RCH; ++r)
        m = fmaxf(m, buf[(size_t)(r0 + r) * NDIM + s]);
    part[(size_t)blockIdx.y * NDIM + s] = m;
}

__global__ __launch_bounds__(256)
void col_max_reduce_kernel(const float* __restrict__ part,
                           float* __restrict__ colmax) {
    const int s = blockIdx.x * 256 + threadIdx.x;
    float m = -INFINITY;
    for (int i = 0; i < NCH; ++i) m = fmaxf(m, part[(size_t)i * NDIM + s]);
    colmax[s] = m;
}

__global__ __launch_bounds__(256)
void col_psum_kernel(const float* __restrict__ sim,
                     const float* __restrict__ colmax,
                     float* __restrict__ part) {
    const int s  = blockIdx.x * 256 + threadIdx.x;
    const int r0 = blockIdx.y * RCH;
    const float cm = colmax[s];
    float sum = 0.f;
    for (int r = 0; r < RCH; ++r)
        sum += expf(sim[(size_t)(r0 + r) * NDIM + s] - cm);
    part[(size_t)blockIdx.y * NDIM + s] = sum;
}

__global__ __launch_bounds__(256)
void col_sum_reduce_kernel(const float* __restrict__ part,
                           float* __restrict__ colsum) {
    const int s = blockIdx.x * 256 + threadIdx.x;
    float sum = 0.f;
    for (int i = 0; i < NCH; ++i) sum += part[(size_t)i * NDIM + s];
    colsum[s] = sum;
}

// ---------------------------------------------------------------------------
// 4) conf = rowSoftmax * colSoftmax, written in place over sim; per-row max
// ---------------------------------------------------------------------------
__global__ __launch_bounds__(256)
void conf_row_kernel(float* __restrict__ buf,
                     const float* __restrict__ rowmax,
                     const float* __restrict__ rowsum,
                     const float* __restrict__ colmax,
                     const float* __restrict__ colsum,
                     float* __restrict__ rowcmax) {
    __shared__ float red[256];
    const int l = blockIdx.x;
    const float rm = rowmax[l];
    const float rs = rowsum[l];
    float* row = buf + (size_t)l * NDIM;

    float m = -INFINITY;
    for (int s = threadIdx.x; s < NDIM; s += 256) {
        const float v = row[s];
        const float c = (expf(v - rm) / rs) * (expf(v - colmax[s]) / colsum[s]);
        row[s] = c;
        m = fmaxf(m, c);
    }
    red[threadIdx.x] = m;
    __syncthreads();
    for (int off = 128; off > 0; off >>= 1) {
        if (threadIdx.x < off)
            red[threadIdx.x] = fmaxf(red[threadIdx.x], red[threadIdx.x + off]);
        __syncthreads();
    }
    if (threadIdx.x == 0) rowcmax[l] = red[0];
}

// ---------------------------------------------------------------------------
// 5) final masked emit (float4 stores)
// ---------------------------------------------------------------------------
__global__ __launch_bounds__(256)
void final_kernel(const float* __restrict__ conf,
                  const float* __restrict__ rowcmax,
                  const float* __restrict__ colcmax,
                  float* __restrict__ out) {
    const size_t idx = (size_t)blockIdx.x * 256 + threadIdx.x;  // float4 index
    const int l  = (int)(idx / (NDIM / 4));
    const int s0 = (int)(idx % (NDIM / 4)) * 4;

    const int i0 = l / HW, j0 = l % HW;
    const bool ml = (i0 >= MARGIN) & (i0 < HW - MARGIN) &
                    (j0 >= MARGIN) & (j0 < HW - MARGIN);
    const float rm = rowcmax[l];

    const float4 v = *(const float4*)(conf + (size_t)l * NDIM + s0);
    float o[4];
    const float cv[4] = {v.x, v.y, v.z, v.w};
#pragma unroll
    for (int t = 0; t < 4; ++t) {
        const int s  = s0 + t;
        const int i1 = s / HW, j1 = s % HW;
        const bool ms = (i1 >= MARGIN) & (i1 < HW - MARGIN) &
                        (j1 >= MARGIN) & (j1 < HW - MARGIN);
        const float c = cv[t];
        const bool ok = ml & ms & (c > THRESH) & (c == rm) & (c == colcmax[s]);
        o[t] = ok ? c : 0.f;
    }
    *(float4*)(out + (size_t)l * NDIM + s0) = make_float4(o[0], o[1], o[2], o[3]);
}

// ---------------------------------------------------------------------------
extern "C" void kernel_launch(void* const* d_in, const int* in_sizes, int n_in,
                              void* d_out, int out_size, void* d_ws, size_t ws_size,
                              hipStream_t stream) {
    (void)in_sizes; (void)n_in; (void)out_size; (void)ws_size;
    const float* x0 = (const float*)d_in[0];
    const float* x1 = (const float*)d_in[1];
    float* out = (float*)d_out;

    // workspace layout
    float* sim     = (float*)d_ws;                       // NDIM*NDIM, becomes conf in place
    float* stats   = sim + (size_t)NDIM * NDIM;
    float* rowmax  = stats;                              // NDIM each
    float* rowsum  = stats + 1 * NDIM;
    float* colmax  = stats + 2 * NDIM;
    float* colsum  = stats + 3 * NDIM;
    float* rowcmax = stats + 4 * NDIM;
    float* colcmax = stats + 5 * NDIM;
    float* part    = stats + 6 * NDIM;                   // NCH * NDIM

    // 1) sim = x0 @ x1^T / 25.6 (WMMA f32)
    gemm_sim_kernel<<<dim3(NDIM / 64, NDIM / 128), 256, 0, stream>>>(x0, x1, sim);

    // 2) row softmax stats
    row_stats_kernel<<<NDIM, 256, 0, stream>>>(sim, rowmax, rowsum);

    // 3) column softmax stats (two-stage, coalesced)
    col_pmax_kernel<<<dim3(NDIM / 256, NCH), 256, 0, stream>>>(sim, part);
    col_max_reduce_kernel<<<NDIM / 256, 256, 0, stream>>>(part, colmax);
    col_psum_kernel<<<dim3(NDIM / 256, NCH), 256, 0, stream>>>(sim, colmax, part);
    col_sum_reduce_kernel<<<NDIM / 256, 256, 0, stream>>>(part, colsum);

    // 4) conf (in place) + per-row conf max
    conf_row_kernel<<<NDIM, 256, 0, stream>>>(sim, rowmax, rowsum, colmax, colsum, rowcmax);

    // 5) per-column conf max (two-stage on conf buffer)
    col_pmax_kernel<<<dim3(NDIM / 256, NCH), 256, 0, stream>>>(sim, part);
    col_max_reduce_kernel<<<NDIM / 256, 256, 0, stream>>>(part, colcmax);

    // 6) final: threshold + margin + mutual top-1
    const int nvec4 = (NDIM / 4) * NDIM;  // 10,240,000 float4 elements
    final_kernel<<<nvec4 / 256, 256, 0, stream>>>(sim, rowcmax, colcmax, out);
}